// PhysicsDiscriminator_51032801411679
// MI455X (gfx1250) — compile-verified
//
#include <hip/hip_runtime.h>

typedef float v2f __attribute__((ext_vector_type(2)));
typedef float v8f __attribute__((ext_vector_type(8)));

#define Bk   32
#define Ek   128
#define Nk   60
#define NTRIk 1830
#define NP   64
#define SP   68   // LDS row stride (floats): 68 banks -> 4-bank rotation per row

// ---------------------------------------------------------------------------
// Complex 16x16 <- X(16x16) * Y(16x16) product accumulated into (pr, pi)
// via V_WMMA_F32_16X16X4_F32, K tiled by 4.
//  A frag (16x4 f32): lanes 0-15 hold M=lane, K = kk+{0,1}; lanes 16-31 K = kk+{2,3}
//  B frag (4x16 f32): lanes 0-15 hold N=lane, K rows kk+{0,1}; lanes 16-31 rows kk+{2,3}
//  pr += Xr*Yr - Xi*Yi ; pi += Xr*Yi + Xi*Yr   (Ai negated at load: f32 WMMA has C-neg only)
// ---------------------------------------------------------------------------
__device__ __forceinline__ void cgemm16(const float* XR, const float* XI, int xoff, int xs,
                                        const float* YR, const float* YI, int yoff, int ys,
                                        v8f& pr, v8f& pi) {
  const int l  = threadIdx.x & 31;
  const int m  = l & 15;
  const int kh = (l >> 4) << 1;
#pragma unroll
  for (int kk = 0; kk < 16; kk += 4) {
    v2f ar, ai, ain, br, bi;
    const int ia = xoff + m * xs + kk + kh;
    ar.x = XR[ia];     ar.y = XR[ia + 1];
    ai.x = XI[ia];     ai.y = XI[ia + 1];
    ain.x = -ai.x;     ain.y = -ai.y;
    const int ib = yoff + (kk + kh) * ys + m;
    br.x = YR[ib];     br.y = YR[ib + ys];
    bi.x = YI[ib];     bi.y = YI[ib + ys];
    pr = __builtin_amdgcn_wmma_f32_16x16x4_f32(false, ar,  false, br, (short)0, pr, false, false);
    pr = __builtin_amdgcn_wmma_f32_16x16x4_f32(false, ain, false, bi, (short)0, pr, false, false);
    pi = __builtin_amdgcn_wmma_f32_16x16x4_f32(false, ar,  false, bi, (short)0, pi, false, false);
    pi = __builtin_amdgcn_wmma_f32_16x16x4_f32(false, ai,  false, br, (short)0, pi, false, false);
  }
}

// C/D tile layout: VGPR r -> row (r + 8*(lane>=16)), col = lane&15
__device__ __forceinline__ void tile_sub(float* R, float* I, int off, int s,
                                         const v8f& pr, const v8f& pi) {
  const int l = threadIdx.x & 31;
  const int col = l & 15, rb = (l >> 4) << 3;
#pragma unroll
  for (int r = 0; r < 8; ++r) {
    const int idx = off + (rb + r) * s + col;
    R[idx] -= pr[r];
    I[idx] -= pi[r];
  }
}
__device__ __forceinline__ void tile_set(float* R, float* I, int off, int s,
                                         const v8f& pr, const v8f& pi) {
  const int l = threadIdx.x & 31;
  const int col = l & 15, rb = (l >> 4) << 3;
#pragma unroll
  for (int r = 0; r < 8; ++r) {
    const int idx = off + (rb + r) * s + col;
    R[idx] = pr[r];
    I[idx] = pi[r];
  }
}
__device__ __forceinline__ void tile_setneg(float* R, float* I, int off, int s,
                                            const v8f& pr, const v8f& pi) {
  const int l = threadIdx.x & 31;
  const int col = l & 15, rb = (l >> 4) << 3;
#pragma unroll
  for (int r = 0; r < 8; ++r) {
    const int idx = off + (rb + r) * s + col;
    R[idx] = -pr[r];
    I[idx] = -pi[r];
  }
}

// ---------------------------------------------------------------------------
// One workgroup per (b, e): build A in LDS, blocked complex Gauss-Jordan
// inversion (NB=16, WMMA trailing updates), then reduce to T and DOS.
// ---------------------------------------------------------------------------
__global__ __launch_bounds__(256) void negf_kernel(
    const float* __restrict__ Hr, const float* __restrict__ Hi,
    const float* __restrict__ GL, const float* __restrict__ GR,
    const float* __restrict__ Eg, float* __restrict__ out) {
  __shared__ float sAr[NP * SP];
  __shared__ float sAi[NP * SP];
  __shared__ float sDr[256];
  __shared__ float sDi[256];
  __shared__ float sGL[NP];
  __shared__ float sGR[NP];
  __shared__ float sRed[16];
  __shared__ float sPR[2];

  const int tid = threadIdx.x;
  const int e = blockIdx.x;
  const int b = blockIdx.y;
  const float Ev = Eg[e];
  const float* hr = Hr + b * NTRIk;
  const float* hi = Hi + b * NTRIk;

  if (tid < NP)           sGL[tid]      = (tid < Nk)      ? GL[tid]      : 0.0f;
  else if (tid < 2 * NP)  sGR[tid - NP] = (tid - NP < Nk) ? GR[tid - NP] : 0.0f;

  // --- build A = E*I - H - diag(sigma); pad rows/cols 60..63 with identity ---
#pragma unroll
  for (int t = 0; t < 16; ++t) {
    const int el = tid + 256 * t;
    const int r = el >> 6, c = el & 63;
    float ar, ai;
    if (r < Nk && c < Nk) {
      if (r == c) {
        const int idx = r * Nk - ((r * (r + 1)) >> 1) + c;
        ar = Ev - hr[idx];                              // H_ii = conj(tri_ii)
        ai = hi[idx] + 0.5f * (GL[r] + GR[r]);          // +i/2 (GL+GR) from -Sigma
      } else if (r < c) {
        const int idx = r * Nk - ((r * (r + 1)) >> 1) + c;
        ar = -hr[idx]; ai = -hi[idx];
      } else {
        const int idx = c * Nk - ((c * (c + 1)) >> 1) + r;
        ar = -hr[idx]; ai = hi[idx];                    // conj fill
      }
    } else {
      ar = (r == c) ? 1.0f : 0.0f; ai = 0.0f;
    }
    sAr[r * SP + c] = ar;
    sAi[r * SP + c] = ai;
  }
  __syncthreads();

  const int w  = tid >> 5;
  const int drw = tid >> 4;   // row within 16x16 diag block
  const int dcw = tid & 15;   // col within 16x16 diag block

  // --- blocked in-place Gauss-Jordan, 4 block steps of NB=16 ---
  for (int k = 0; k < 4; ++k) {
    const int k16 = k << 4;
    int ob[3];
    { int q = 0;
#pragma unroll
      for (int x = 0; x < 4; ++x) if (x != k) ob[q++] = x; }

    // Phase A: Dinv = inv(A[k][k]) (cooperative unblocked complex GJ, no pivoting)
    sDr[tid] = sAr[(k16 + drw) * SP + k16 + dcw];
    sDi[tid] = sAi[(k16 + drw) * SP + k16 + dcw];
    for (int p = 0; p < 16; ++p) {
      __syncthreads();
      if (tid == 0) {
        const float xr = sDr[p * 16 + p], xi = sDi[p * 16 + p];
        const float inv = 1.0f / (xr * xr + xi * xi);
        sPR[0] = xr * inv; sPR[1] = -xi * inv;
      }
      __syncthreads();
      const float rr = sPR[0], ri = sPR[1];
      if (drw == p) {                                    // scale pivot row
        float vr, vi;
        if (dcw == p) { vr = rr; vi = ri; }
        else {
          const float xr = sDr[p * 16 + dcw], xi = sDi[p * 16 + dcw];
          vr = xr * rr - xi * ri; vi = xr * ri + xi * rr;
        }
        sDr[p * 16 + dcw] = vr; sDi[p * 16 + dcw] = vi;
      }
      __syncthreads();
      float vr = 0.0f, vi = 0.0f;
      if (drw != p) {                                    // eliminate
        const float fr = sDr[drw * 16 + p], fi = sDi[drw * 16 + p];
        if (dcw == p) { vr = -(fr * rr - fi * ri); vi = -(fr * ri + fi * rr); }
        else {
          const float xr = sDr[p * 16 + dcw], xi = sDi[p * 16 + dcw];
          vr = sDr[drw * 16 + dcw] - (fr * xr - fi * xi);
          vi = sDi[drw * 16 + dcw] - (fr * xi + fi * xr);
        }
      }
      __syncthreads();
      if (drw != p) { sDr[drw * 16 + dcw] = vr; sDi[drw * 16 + dcw] = vi; }
    }
    __syncthreads();

    // Phase B: pivot row  A[k][j] = Dinv * A[k][j]  (waves 0..2); wave 3: A[k][k] = Dinv
    if (w < 3) {
      const int j16 = ob[w] << 4;
      v8f pr = {0,0,0,0,0,0,0,0}, pi = {0,0,0,0,0,0,0,0};
      cgemm16(sDr, sDi, 0, 16, sAr, sAi, k16 * SP + j16, SP, pr, pi);
      tile_set(sAr, sAi, k16 * SP + j16, SP, pr, pi);
    } else if (w == 3) {
      const int l = tid & 31;
#pragma unroll
      for (int t = 0; t < 8; ++t) {
        const int el = l * 8 + t;
        sAr[(k16 + (el >> 4)) * SP + k16 + (el & 15)] = sDr[el];
        sAi[(k16 + (el >> 4)) * SP + k16 + (el & 15)] = sDi[el];
      }
    }
    __syncthreads();

    // Phase C: trailing update  A[i][j] -= A[i][k] * A[k][j]  (9 tiles over 8 waves)
    for (int t = w; t < 9; t += 8) {
      const int i16 = ob[t / 3] << 4;
      const int j16 = ob[t % 3] << 4;
      v8f pr = {0,0,0,0,0,0,0,0}, pi = {0,0,0,0,0,0,0,0};
      cgemm16(sAr, sAi, i16 * SP + k16, SP, sAr, sAi, k16 * SP + j16, SP, pr, pi);
      tile_sub(sAr, sAi, i16 * SP + j16, SP, pr, pi);
    }
    __syncthreads();

    // Phase D: pivot column  A[i][k] = -A[i][k] * Dinv  (waves 0..2)
    if (w < 3) {
      const int i16 = ob[w] << 4;
      v8f pr = {0,0,0,0,0,0,0,0}, pi = {0,0,0,0,0,0,0,0};
      cgemm16(sAr, sAi, i16 * SP + k16, SP, sDr, sDi, 0, 16, pr, pi);
      tile_setneg(sAr, sAi, i16 * SP + k16, SP, pr, pi);
    }
    __syncthreads();
  }

  // --- reductions: T = sum GL_i GR_j |Gr_ij|^2 ; DOS = -sum Im Gr_nn ---
  float tacc = 0.0f, dacc = 0.0f;
#pragma unroll
  for (int t = 0; t < 16; ++t) {
    const int el = tid + 256 * t;
    const int r = el >> 6, c = el & 63;
    const float gr = sAr[r * SP + c], gi = sAi[r * SP + c];
    tacc += sGL[r] * sGR[c] * (gr * gr + gi * gi);   // padded gammas are 0
    if (r == c && r < Nk) dacc -= gi;
  }
#pragma unroll
  for (int off = 16; off > 0; off >>= 1) {
    tacc += __shfl_xor(tacc, off, 32);
    dacc += __shfl_xor(dacc, off, 32);
  }
  if ((tid & 31) == 0) { sRed[w] = tacc; sRed[8 + w] = dacc; }
  __syncthreads();
  if (tid == 0) {
    float T = 0.0f, D = 0.0f;
#pragma unroll
    for (int q = 0; q < 8; ++q) { T += sRed[q]; D += sRed[8 + q]; }
    out[b * Ek + e] = T;                   // T  [B, E]
    out[Bk * Ek + b * Ek + e] = D;         // DOS[B, E]
  }
}

extern "C" void kernel_launch(void* const* d_in, const int* in_sizes, int n_in,
                              void* d_out, int out_size, void* d_ws, size_t ws_size,
                              hipStream_t stream) {
  const float* Hr = (const float*)d_in[0];
  const float* Hi = (const float*)d_in[1];
  const float* GL = (const float*)d_in[2];
  const float* GR = (const float*)d_in[3];
  const float* Eg = (const float*)d_in[4];
  dim3 grid(Ek, Bk);
  negf_kernel<<<grid, dim3(256), 0, stream>>>(Hr, Hi, GL, GR, Eg, (float*)d_out);
}